// ALIGNNStyle_60181081752149
// MI455X (gfx1250) — compile-verified
//
#include <hip/hip_runtime.h>
#include <math.h>

#define N_NODES 20000
#define N_EDGES 600000
#define ET (N_EDGES + N_NODES)   // edges + self loops = 620000
#define G_GR 128
#define NFEAT 16
#define HIDD 128
#define LAYERS 4
#define HEADS 4
#define CHD 32                    // per-head channels
#define KMAX 128

typedef __attribute__((ext_vector_type(2))) float v2f;
typedef __attribute__((ext_vector_type(8))) float v8f;

// ---------------- float atomic max (bit-trick; init to -inf) ----------------
__device__ __forceinline__ void atomicMaxFloat(float* addr, float val) {
  if (val >= 0.0f)
    atomicMax((int*)addr, __float_as_int(val));
  else
    atomicMin((unsigned int*)addr, __float_as_uint(val));
}

// ---------------- WMMA f32 GEMM: C[M,Ncols] = A[M,K] @ B[K,Ncols] -----------
// One block (256 thr = 8 waves) computes a 16-row x 128-col output strip:
// wave w owns the 16x16 tile at columns [16w, 16w+16). The 16xK A tile is a
// contiguous block of A (full-K extent), staged once into LDS with coalesced
// float4 copies, then fed to V_WMMA_F32_16X16X4_F32 via aligned ds_load_b64.
// A-layout (ISA 7.12.2): lanes 0-15 hold K={k0,k0+1}, lanes 16-31 K={k0+2,k0+3}.
// C-layout: VGPR r -> row r (lanes 0-15) / row r+8 (lanes 16-31), col = lane&15.
__global__ void gemm_wmma_f32(const float* __restrict__ A,
                              const float* __restrict__ B,
                              const float* __restrict__ bias,
                              float* __restrict__ C,
                              int K, int Ncols, int do_relu) {
  __shared__ float sA[16 * KMAX];

  const int tid = threadIdx.x;
  const int wave = tid >> 5;
  const int lane = tid & 31;
  const int row0 = blockIdx.x << 4;

  // cooperative coalesced stage of the contiguous 16xK A tile
  {
    const float4* Asrc = (const float4*)(A + (size_t)row0 * K);
    float4* sA4 = (float4*)sA;
    const int n4 = (16 * K) >> 2;
    for (int i = tid; i < n4; i += 256) sA4[i] = Asrc[i];
  }
  __syncthreads();

  const int col0 = wave << 4;                  // 8 waves cover 128 cols
  const int halfSel = lane >> 4;               // 0: K pair {0,1}; 1: {2,3}
  const int m16 = lane & 15;
  const float* aRow = sA + m16 * K;

  v8f c = {};
  for (int k0 = 0; k0 < K; k0 += 4) {
    const int ka = k0 + (halfSel << 1);
    v2f a = *(const v2f*)(aRow + ka);          // ds_load_b64 (8B aligned)
    v2f b;
    b.x = B[(size_t)ka * Ncols + col0 + m16];
    b.y = B[(size_t)(ka + 1) * Ncols + col0 + m16];
    c = __builtin_amdgcn_wmma_f32_16x16x4_f32(false, a, false, b, (short)0, c,
                                              false, false);
  }

  const int orow0 = row0 + (halfSel ? 8 : 0);
  const int ocol = col0 + m16;
  const float bv = bias ? bias[ocol] : 0.0f;
#pragma unroll
  for (int r = 0; r < 8; r++) {
    float v = c[r] + bv;
    if (do_relu) v = v > 0.0f ? v : 0.0f;
    C[(size_t)(orow0 + r) * Ncols + ocol] = v;
  }
}

// ------------- per-node prep: attention scores + init accumulators ----------
__global__ void node_prep(const float* __restrict__ hw,
                          const float* __restrict__ a_src,
                          const float* __restrict__ a_dst,
                          float* __restrict__ s_src, float* __restrict__ s_dst,
                          float* __restrict__ mmax, float* __restrict__ denom,
                          float* __restrict__ agg) {
  const int t = blockIdx.x * blockDim.x + threadIdx.x;
  if (t >= N_NODES * HEADS) return;
  const int n = t >> 2;
  const int h = t & 3;
  const float4* hp = (const float4*)(hw + (size_t)n * HIDD + h * CHD);
  const float4* as = (const float4*)(a_src + h * CHD);
  const float4* ad = (const float4*)(a_dst + h * CHD);
  float4* ag = (float4*)(agg + (size_t)n * HIDD + h * CHD);
  float ss = 0.0f, sd = 0.0f;
  const float4 z = make_float4(0.f, 0.f, 0.f, 0.f);
#pragma unroll
  for (int i = 0; i < CHD / 4; i++) {
    float4 v = hp[i], s = as[i], d = ad[i];
    ss += v.x * s.x + v.y * s.y + v.z * s.z + v.w * s.w;
    sd += v.x * d.x + v.y * d.y + v.z * d.z + v.w * d.w;
    ag[i] = z;                                  // zero aggregate buffer
  }
  s_src[t] = ss;
  s_dst[t] = sd;
  mmax[t] = -INFINITY;
  denom[t] = 0.0f;
}

__device__ __forceinline__ void edge_sd(const int* __restrict__ ei, int e,
                                        int& s, int& d) {
  if (e < N_EDGES) { s = ei[e]; d = ei[N_EDGES + e]; }
  else             { s = e - N_EDGES; d = s; }   // self loop
}

// ---------------- pass 1: leaky-relu logits + segment max --------------------
__global__ void edge_logits(const int* __restrict__ ei,
                            const float* __restrict__ s_src,
                            const float* __restrict__ s_dst,
                            float* __restrict__ exb, float* __restrict__ mmax) {
  const int t = blockIdx.x * blockDim.x + threadIdx.x;
  if (t >= ET * HEADS) return;
  const int e = t >> 2, h = t & 3;
  int s, d; edge_sd(ei, e, s, d);
  float x = s_src[s * HEADS + h] + s_dst[d * HEADS + h];
  x = x > 0.0f ? x : 0.2f * x;
  exb[t] = x;
  atomicMaxFloat(&mmax[d * HEADS + h], x);
}

// ---------------- pass 2: exp + segment sum ---------------------------------
__global__ void edge_exp(const int* __restrict__ ei, float* __restrict__ exb,
                         const float* __restrict__ mmax,
                         float* __restrict__ denom) {
  const int t = blockIdx.x * blockDim.x + threadIdx.x;
  if (t >= ET * HEADS) return;
  const int e = t >> 2, h = t & 3;
  int s, d; edge_sd(ei, e, s, d);
  const float ev = __expf(exb[t] - mmax[d * HEADS + h]);
  exb[t] = ev;
  atomicAdd(&denom[d * HEADS + h], ev);
}

// ---------------- pass 3: weighted message scatter ---------------------------
__global__ void edge_msg(const int* __restrict__ ei,
                         const float* __restrict__ exb,
                         const float* __restrict__ denom,
                         const float* __restrict__ hw, float* __restrict__ agg) {
  const int t = blockIdx.x * blockDim.x + threadIdx.x;
  if (t >= ET * HEADS) return;
  const int e = t >> 2, h = t & 3;
  int s, d; edge_sd(ei, e, s, d);
  const float alpha = exb[t] / (denom[d * HEADS + h] + 1e-16f);
  const float4* hs = (const float4*)(hw + (size_t)s * HIDD + h * CHD);
  float* ag = agg + (size_t)d * HIDD + h * CHD;
#pragma unroll
  for (int i = 0; i < CHD / 4; i++) {
    float4 v = hs[i];
    atomicAdd(ag + 4 * i + 0, v.x * alpha);
    atomicAdd(ag + 4 * i + 1, v.y * alpha);
    atomicAdd(ag + 4 * i + 2, v.z * alpha);
    atomicAdd(ag + 4 * i + 3, v.w * alpha);
  }
}

// ---------------- bias + batchnorm + residual relu ---------------------------
__global__ void post_layer(const float* __restrict__ agg,
                           const float* __restrict__ gb,
                           const float* __restrict__ gam,
                           const float* __restrict__ bet,
                           const float* __restrict__ mu,
                           const float* __restrict__ var,
                           float* __restrict__ h) {
  const int t = blockIdx.x * blockDim.x + threadIdx.x;
  if (t >= N_NODES * HIDD) return;
  const int ch = t & (HIDD - 1);
  float hn = agg[t] + gb[ch];
  hn = (hn - mu[ch]) * (gam[ch] * rsqrtf(var[ch] + 1e-5f)) + bet[ch];
  h[t] = h[t] + (hn > 0.0f ? hn : 0.0f);
}

// ---------------- pooling ----------------------------------------------------
__global__ void pool_zero(float* __restrict__ gsum, float* __restrict__ cnt) {
  const int t = blockIdx.x * blockDim.x + threadIdx.x;
  if (t < G_GR * HIDD) gsum[t] = 0.0f;
  if (t < G_GR) cnt[t] = 0.0f;
}

__global__ void pool_sum(const float* __restrict__ h, const int* __restrict__ batch,
                         float* __restrict__ gsum, float* __restrict__ cnt) {
  const int t = blockIdx.x * blockDim.x + threadIdx.x;
  if (t >= N_NODES * HIDD) return;
  const int n = t >> 7;
  const int ch = t & (HIDD - 1);
  const int b = batch[n];
  atomicAdd(&gsum[b * HIDD + ch], h[t]);
  if (ch == 0) atomicAdd(&cnt[b], 1.0f);
}

// ---------------- MLP head: 128 -> 64 -> 32 -> 1 -----------------------------
__global__ void head_mlp(const float* __restrict__ gsum, const float* __restrict__ cnt,
                         const float* __restrict__ w1, const float* __restrict__ b1,
                         const float* __restrict__ w2, const float* __restrict__ b2,
                         const float* __restrict__ wg, const float* __restrict__ bg,
                         float* __restrict__ out) {
  const int g = threadIdx.x;
  if (g >= G_GR) return;
  const float inv = 1.0f / fmaxf(cnt[g], 1.0f);
  float a1[64];
#pragma unroll
  for (int j = 0; j < 64; j++) a1[j] = b1[j];
  for (int k = 0; k < HIDD; k++) {
    const float x = gsum[g * HIDD + k] * inv;
#pragma unroll
    for (int j = 0; j < 64; j++) a1[j] += x * w1[k * 64 + j];
  }
  float a2[32];
#pragma unroll
  for (int j = 0; j < 32; j++) a2[j] = b2[j];
  for (int k = 0; k < 64; k++) {
    const float x = fmaxf(a1[k], 0.0f);
#pragma unroll
    for (int j = 0; j < 32; j++) a2[j] += x * w2[k * 32 + j];
  }
  float o = bg[0];
#pragma unroll
  for (int k = 0; k < 32; k++) o += fmaxf(a2[k], 0.0f) * wg[k];
  out[g] = o;
}

// ---------------- launch -----------------------------------------------------
extern "C" void kernel_launch(void* const* d_in, const int* in_sizes, int n_in,
                              void* d_out, int out_size, void* d_ws, size_t ws_size,
                              hipStream_t stream) {
  const float* x        = (const float*)d_in[0];
  const int*   ei       = (const int*)  d_in[1];
  const int*   batch    = (const int*)  d_in[2];
  const float* emb_w    = (const float*)d_in[3];
  const float* emb_b    = (const float*)d_in[4];
  const float* gat_w    = (const float*)d_in[5];
  const float* att_src  = (const float*)d_in[6];
  const float* att_dst  = (const float*)d_in[7];
  const float* gat_b    = (const float*)d_in[8];
  const float* bn_gamma = (const float*)d_in[9];
  const float* bn_beta  = (const float*)d_in[10];
  const float* bn_mean  = (const float*)d_in[11];
  const float* bn_var   = (const float*)d_in[12];
  const float* fc1_w    = (const float*)d_in[13];
  const float* fc1_b    = (const float*)d_in[14];
  const float* fc2_w    = (const float*)d_in[15];
  const float* fc2_b    = (const float*)d_in[16];
  const float* bg_w     = (const float*)d_in[17];
  const float* bg_b     = (const float*)d_in[18];
  float* out = (float*)d_out;

  float* ws    = (float*)d_ws;
  float* h     = ws; ws += (size_t)N_NODES * HIDD;
  float* hw    = ws; ws += (size_t)N_NODES * HIDD;
  float* agg   = ws; ws += (size_t)N_NODES * HIDD;
  float* ssrc  = ws; ws += (size_t)N_NODES * HEADS;
  float* sdst  = ws; ws += (size_t)N_NODES * HEADS;
  float* mmax  = ws; ws += (size_t)N_NODES * HEADS;
  float* denom = ws; ws += (size_t)N_NODES * HEADS;
  float* exb   = ws; ws += (size_t)ET * HEADS;
  float* gsum  = ws; ws += (size_t)G_GR * HIDD;
  float* cnt   = ws; ws += (size_t)G_GR;

  const int gblocks  = N_NODES / 16;   // 1250 blocks, one 16-row strip each
  const int ethreads = ET * HEADS;
  const int nthreads = N_NODES * HIDD;

  // embedding: h = relu(x @ emb_w + emb_b)
  gemm_wmma_f32<<<gblocks, 256, 0, stream>>>(x, emb_w, emb_b, h,
                                             NFEAT, HIDD, 1);

  for (int l = 0; l < LAYERS; l++) {
    gemm_wmma_f32<<<gblocks, 256, 0, stream>>>(h, gat_w + (size_t)l * HIDD * HIDD,
                                               nullptr, hw, HIDD, HIDD, 0);
    node_prep<<<(N_NODES * HEADS + 255) / 256, 256, 0, stream>>>(
        hw, att_src + (size_t)l * HEADS * CHD, att_dst + (size_t)l * HEADS * CHD,
        ssrc, sdst, mmax, denom, agg);
    edge_logits<<<(ethreads + 255) / 256, 256, 0, stream>>>(ei, ssrc, sdst, exb, mmax);
    edge_exp<<<(ethreads + 255) / 256, 256, 0, stream>>>(ei, exb, mmax, denom);
    edge_msg<<<(ethreads + 255) / 256, 256, 0, stream>>>(ei, exb, denom, hw, agg);
    post_layer<<<(nthreads + 255) / 256, 256, 0, stream>>>(
        agg, gat_b + (size_t)l * HIDD, bn_gamma + (size_t)l * HIDD,
        bn_beta + (size_t)l * HIDD, bn_mean + (size_t)l * HIDD,
        bn_var + (size_t)l * HIDD, h);
  }

  pool_zero<<<(G_GR * HIDD + 255) / 256, 256, 0, stream>>>(gsum, cnt);
  pool_sum<<<(nthreads + 255) / 256, 256, 0, stream>>>(h, batch, gsum, cnt);
  head_mlp<<<1, 128, 0, stream>>>(gsum, cnt, fc1_w, fc1_b, fc2_w, fc2_b,
                                  bg_w, bg_b, out);
}